// GatedGCNGraphGymLayer_26182120636870
// MI455X (gfx1250) — compile-verified
//
#include <hip/hip_runtime.h>
#include <cstdint>
#include <cstddef>

// GatedGCN layer for MI455X (gfx1250, wave32, WMMA).
// GEMMs use V_WMMA_F32_16X16X4_F32 (fp32 A/B, fp32 accum) to match the fp32 reference.
// Edge pass is fully fused: Ce tile (WMMA) + Dx[dst] + Ex[src] gathers (L2-resident)
// + sigmoid + scatter-sum atomics + raw e_ij store. BN done as column reduction + in-place pass.

#define NN 50000
#define NE 800000
#define DD 128
#define LDS_PITCH 132   // 132 mod 64 = 4 -> conflict-free column reads across 16 rows

typedef __attribute__((ext_vector_type(2))) float v2f;
typedef __attribute__((ext_vector_type(8))) float v8f;

// ---------------------------------------------------------------------------
// One 16x16 output tile, K=128, via 32x V_WMMA_F32_16X16X4_F32.
// A: 16x128 tile in LDS (row-major, pitch LDS_PITCH).
// B: 128x128 row-major weight in global (L2-resident, only 64KB).
// ISA layouts (CDNA5 ISA):
//   A 16x4:  lanes 0-15 hold M=lane, VGPR0/1 = K0/K1; lanes 16-31: K2/K3.
//   B 4x16:  rows striped across lanes within a VGPR (mirror of C/D).
//   C/D 16x16: lane n = N, vgpr r = M (lanes 16-31: M = r+8).
// ---------------------------------------------------------------------------
__device__ __forceinline__ v8f wmma16x16x128(const float* __restrict__ tile,
                                             const float* __restrict__ W,
                                             int nbase, int lane) {
  const int half = lane >> 4;      // 0: lanes 0-15, 1: lanes 16-31
  const int mn   = lane & 15;      // M for A-frag, N for B-frag
  v8f c = {0.f, 0.f, 0.f, 0.f, 0.f, 0.f, 0.f, 0.f};
  const float* arow = tile + mn * LDS_PITCH + 2 * half;         // A[m][k + 2*half]
  const float* bcol = W + (size_t)(2 * half) * DD + nbase + mn; // W[k + 2*half][nbase+n]
  #pragma unroll 8
  for (int k = 0; k < DD; k += 4) {
    v2f a, b;
    a.x = arow[k];
    a.y = arow[k + 1];
    b.x = bcol[(size_t)k * DD];
    b.y = bcol[(size_t)(k + 1) * DD];
    c = __builtin_amdgcn_wmma_f32_16x16x4_f32(false, a, false, b, (short)0, c,
                                              false, false);
  }
  return c;
}

// Stage a 16x128 fp32 tile (row-major) into LDS with pitch LDS_PITCH. 256 threads.
__device__ __forceinline__ void stage16x128(const float* __restrict__ src,
                                            float* __restrict__ tile, int tid) {
  const int row = tid >> 4;          // 0..15
  const int cg  = (tid & 15) * 8;    // 0,8,...,120
  const float4* g = (const float4*)(src + (size_t)row * DD + cg);
  float4 v0 = g[0], v1 = g[1];
  float* t = tile + row * LDS_PITCH + cg;
  t[0] = v0.x; t[1] = v0.y; t[2] = v0.z; t[3] = v0.w;
  t[4] = v1.x; t[5] = v1.y; t[6] = v1.z; t[7] = v1.w;
}

// ---------------------------------------------------------------------------
// Kernel 1: node GEMMs. One 16-node row tile per block (x staged in LDS once,
// reused for all four weight matrices). 8 waves -> 8 column slices of 16.
// ---------------------------------------------------------------------------
__global__ __launch_bounds__(256) void node_gemm4_kernel(
    const float* __restrict__ x,
    const float* __restrict__ WA, const float* __restrict__ bA,
    const float* __restrict__ WB, const float* __restrict__ bB,
    const float* __restrict__ WD, const float* __restrict__ bD,
    const float* __restrict__ WE, const float* __restrict__ bE,
    float* __restrict__ Ax, float* __restrict__ Bx,
    float* __restrict__ Dx, float* __restrict__ Ex) {
  __shared__ float tile[16 * LDS_PITCH];
  const int tid  = threadIdx.x;
  const int lane = tid & 31;
  const int wave = tid >> 5;
  const int rowbase = blockIdx.x * 16;
  stage16x128(x + (size_t)rowbase * DD, tile, tid);
  __syncthreads();

  const int nbase = wave * 16;
  const int half  = lane >> 4;
  const int col   = nbase + (lane & 15);
  const int r0    = rowbase + 8 * half;

  v8f c;
  c = wmma16x16x128(tile, WA, nbase, lane);
  { const float b = bA[col];
    #pragma unroll
    for (int r = 0; r < 8; ++r) Ax[(size_t)(r0 + r) * DD + col] = c[r] + b; }
  c = wmma16x16x128(tile, WB, nbase, lane);
  { const float b = bB[col];
    #pragma unroll
    for (int r = 0; r < 8; ++r) Bx[(size_t)(r0 + r) * DD + col] = c[r] + b; }
  c = wmma16x16x128(tile, WD, nbase, lane);
  { const float b = bD[col];
    #pragma unroll
    for (int r = 0; r < 8; ++r) Dx[(size_t)(r0 + r) * DD + col] = c[r] + b; }
  c = wmma16x16x128(tile, WE, nbase, lane);
  { const float b = bE[col];
    #pragma unroll
    for (int r = 0; r < 8; ++r) Ex[(size_t)(r0 + r) * DD + col] = c[r] + b; }
}

// ---------------------------------------------------------------------------
// Kernel 2: fused edge pass. Per 16-edge tile: Ce via WMMA, add gathered
// Dx[dst]/Ex[src] (L2-resident), sigmoid, store raw e_ij to d_out, and do the
// normalized scatter-sum via fp32 atomics into num/den (L2-resident).
// ---------------------------------------------------------------------------
__global__ __launch_bounds__(256) void edge_fused_kernel(
    const float* __restrict__ e,
    const float* __restrict__ WC, const float* __restrict__ bC,
    const float* __restrict__ Bx, const float* __restrict__ Dx,
    const float* __restrict__ Ex, const int* __restrict__ ei,
    float* __restrict__ e_out, float* __restrict__ num, float* __restrict__ den) {
  __shared__ float tile[16 * LDS_PITCH];
  const int tid  = threadIdx.x;
  const int lane = tid & 31;
  const int wave = tid >> 5;
  const int ebase = blockIdx.x * 16;
  stage16x128(e + (size_t)ebase * DD, tile, tid);
  __syncthreads();

  const int nbase = wave * 16;
  const int half  = lane >> 4;
  const int col   = nbase + (lane & 15);

  v8f c = wmma16x16x128(tile, WC, nbase, lane);
  const float bias = bC[col];

  #pragma unroll
  for (int r = 0; r < 8; ++r) {
    const int edge = ebase + 8 * half + r;        // M index of this c element
    const int s = ei[edge];                       // source j
    const int d = ei[NE + edge];                  // target i
    const float eij = c[r] + bias + Dx[(size_t)d * DD + col]
                             + Ex[(size_t)s * DD + col];
    e_out[(size_t)edge * DD + col] = eij;         // raw e_ij; BN'd later in place
    const float sg = 1.0f / (1.0f + __expf(-eij));
    atomicAdd(den + (size_t)d * DD + col, sg);
    atomicAdd(num + (size_t)d * DD + col, sg * Bx[(size_t)s * DD + col]);
  }
}

// Kernel 3: x_out_raw = Ax + num / (den + eps)
__global__ __launch_bounds__(256) void x_combine_kernel(
    const float* __restrict__ Ax, const float* __restrict__ num,
    const float* __restrict__ den, float* __restrict__ xo) {
  const size_t i = (size_t)blockIdx.x * blockDim.x + threadIdx.x;
  if (i < (size_t)NN * DD) xo[i] = Ax[i] + num[i] / (den[i] + 1e-6f);
}

// Kernel 4: per-column sum / sum-of-squares (block-local combine, then atomics).
__global__ __launch_bounds__(256) void colstats_kernel(
    const float* __restrict__ X, int rows,
    float* __restrict__ sum, float* __restrict__ sumsq) {
  __shared__ float sh_s[DD], sh_q[DD];
  const int col = threadIdx.x & (DD - 1);
  const int sub = threadIdx.x >> 7;   // two rows in flight per block
  float s = 0.f, q = 0.f;
  for (int r = blockIdx.x * 2 + sub; r < rows; r += gridDim.x * 2) {
    const float v = X[(size_t)r * DD + col];
    s += v; q += v * v;
  }
  if (sub) { sh_s[col] = s; sh_q[col] = q; }
  __syncthreads();
  if (!sub) {
    atomicAdd(sum + col, s + sh_s[col]);
    atomicAdd(sumsq + col, q + sh_q[col]);
  }
}

// Kernel 5: in-place BatchNorm (biased var) + ReLU.
__global__ __launch_bounds__(256) void bn_relu_kernel(
    float* __restrict__ X, const float* __restrict__ sum,
    const float* __restrict__ sumsq, const float* __restrict__ gamma,
    const float* __restrict__ beta, int rows) {
  const size_t total = (size_t)rows * DD;
  const float inv = 1.0f / (float)rows;
  for (size_t i = (size_t)blockIdx.x * blockDim.x + threadIdx.x; i < total;
       i += (size_t)gridDim.x * blockDim.x) {
    const int col = (int)(i & (DD - 1));
    const float m = sum[col] * inv;
    const float v = sumsq[col] * inv - m * m;
    const float y = (X[i] - m) * rsqrtf(v + 1e-5f) * gamma[col] + beta[col];
    X[i] = fmaxf(y, 0.f);
  }
}

__global__ __launch_bounds__(256) void zero_kernel(float* __restrict__ p, size_t n) {
  for (size_t i = (size_t)blockIdx.x * blockDim.x + threadIdx.x; i < n;
       i += (size_t)gridDim.x * blockDim.x)
    p[i] = 0.f;
}

// ---------------------------------------------------------------------------
extern "C" void kernel_launch(void* const* d_in, const int* in_sizes, int n_in,
                              void* d_out, int out_size, void* d_ws, size_t ws_size,
                              hipStream_t stream) {
  (void)in_sizes; (void)n_in; (void)out_size; (void)ws_size;
  const float* x   = (const float*)d_in[0];
  const float* e   = (const float*)d_in[1];
  const float* WA  = (const float*)d_in[2];
  const float* bA  = (const float*)d_in[3];
  const float* WB  = (const float*)d_in[4];
  const float* bB  = (const float*)d_in[5];
  const float* WC  = (const float*)d_in[6];
  const float* bC  = (const float*)d_in[7];
  const float* WD  = (const float*)d_in[8];
  const float* bD  = (const float*)d_in[9];
  const float* WE  = (const float*)d_in[10];
  const float* bE  = (const float*)d_in[11];
  const float* gx  = (const float*)d_in[12];
  const float* btx = (const float*)d_in[13];
  const float* ge  = (const float*)d_in[14];
  const float* bte = (const float*)d_in[15];
  const int*   ei  = (const int*)d_in[16];

  const size_t szN = (size_t)NN * DD;      // 6.4M floats per node matrix
  float* ws  = (float*)d_ws;
  float* Ax  = ws;                          // node scratch: 6 * 25.6 MB (L2-resident)
  float* Bx  = Ax + szN;
  float* Dx  = Bx + szN;
  float* Ex  = Dx + szN;
  float* num = Ex + szN;
  float* den = num + szN;
  float* xsum = den + szN;                  // 4 x 128 floats of BN stats
  float* xsq  = xsum + DD;
  float* esum = xsq + DD;
  float* esq  = esum + DD;

  float* xo = (float*)d_out;                // x_out: 50000 x 128
  float* eo = xo + szN;                     // e_out: 800000 x 128

  // num, den, and the 4 stats vectors are contiguous -> single zero pass.
  zero_kernel<<<4096, 256, 0, stream>>>(num, 2 * szN + 4 * DD);

  node_gemm4_kernel<<<NN / 16, 256, 0, stream>>>(x, WA, bA, WB, bB, WD, bD,
                                                 WE, bE, Ax, Bx, Dx, Ex);

  edge_fused_kernel<<<NE / 16, 256, 0, stream>>>(e, WC, bC, Bx, Dx, Ex, ei,
                                                 eo, num, den);

  x_combine_kernel<<<(int)(szN / 256), 256, 0, stream>>>(Ax, num, den, xo);

  colstats_kernel<<<256, 256, 0, stream>>>(xo, NN, xsum, xsq);
  colstats_kernel<<<2048, 256, 0, stream>>>(eo, NE, esum, esq);

  bn_relu_kernel<<<2048, 256, 0, stream>>>(xo, xsum, xsq, gx, btx, NN);
  bn_relu_kernel<<<8192, 256, 0, stream>>>(eo, esum, esq, ge, bte, NE);
}